// InformationGainObjectDetection_38826504356400
// MI455X (gfx1250) — compile-verified
//
#include <hip/hip_runtime.h>
#include <stdint.h>

// Problem constants (match reference setup_inputs / _forward)
#define N_CUR   128
#define N_PREV  160
#define K_RECT  (N_CUR + N_CUR + N_PREV)   // 416 rectangles
#define HS      1080                        // H/SUB
#define WSUB    1920                        // W/SUB
#define W_FULL  3840
#define RECT_DW 8                           // dwords per rect entry (32B)
#define N_CHUNK ((K_RECT * RECT_DW * 4) / 16)  // 832 x 16-byte chunks

// Paint tile geometry (half-res): 64 cols x 32 rows per block of 256 threads.
#define TILE_W  64
#define TILE_H  32
#define ROWS_PER_THREAD 4      // 8 thread-rows x 4 passes = 32 rows
#define ROW_STRIDE      8

// ---------------------------------------------------------------------------
// Kernel 1: single small block. Computes IoU matrix logic + rect table.
// Rect entry (8 dwords): x1,y1,x2,y2 (int, half-res), valbits (f32), pad[3].
// Inactive rects are written as degenerate (0,0,0,0,val=0) -> never paint.
// ---------------------------------------------------------------------------
__global__ __launch_bounds__(256) void ig_prep_kernel(
    const float* __restrict__ boxes, const float* __restrict__ scores,
    const float* __restrict__ flags, const float* __restrict__ boxes_prev,
    const float* __restrict__ scores_prev, const float* __restrict__ flags_prev,
    const int* __restrict__ ids, const int* __restrict__ ids_prev,
    int* __restrict__ rects)
{
    __shared__ int   bp[N_PREV][4];
    __shared__ float areap[N_PREV];
    __shared__ int   idsp[N_PREV];
    __shared__ int   matched[N_PREV];
    __shared__ int   pj_s[N_CUR];
    __shared__ float pv_s[N_CUR];
    __shared__ int   pact_s[N_CUR];

    const int t = threadIdx.x;

    // Load + floor-subsample previous boxes, areas, ids; clear matched.
    for (int j = t; j < N_PREV; j += 256) {
        int b0 = (int)floorf(boxes_prev[j * 4 + 0] / 2.0f);
        int b1 = (int)floorf(boxes_prev[j * 4 + 1] / 2.0f);
        int b2 = (int)floorf(boxes_prev[j * 4 + 2] / 2.0f);
        int b3 = (int)floorf(boxes_prev[j * 4 + 3] / 2.0f);
        bp[j][0] = b0; bp[j][1] = b1; bp[j][2] = b2; bp[j][3] = b3;
        areap[j] = (float)(b2 - b0) * (float)(b3 - b1);
        idsp[j] = ids_prev[j];
        matched[j] = 0;
    }
    __syncthreads();

    if (t < N_CUR) {
        const int c0 = (int)floorf(boxes[t * 4 + 0] / 2.0f);
        const int c1 = (int)floorf(boxes[t * 4 + 1] / 2.0f);
        const int c2 = (int)floorf(boxes[t * 4 + 2] / 2.0f);
        const int c3 = (int)floorf(boxes[t * 4 + 3] / 2.0f);
        const float area = (float)(c2 - c0) * (float)(c3 - c1);
        const int myid = ids[t];

        int   midx = 0;  bool hasm = false;
        float ioum = 0.0f, iou0 = 0.0f;
        float best = -2.0f; int bidx = 0;   // -2 init => first-index argmax like jnp

        for (int j = 0; j < N_PREV; ++j) {
            float ix1 = fmaxf((float)c0, (float)bp[j][0]);
            float iy1 = fmaxf((float)c1, (float)bp[j][1]);
            float ix2 = fminf((float)c2, (float)bp[j][2]);
            float iy2 = fminf((float)c3, (float)bp[j][3]);
            float inter = fmaxf(ix2 - ix1, 0.0f) * fmaxf(iy2 - iy1, 0.0f);
            float iou = inter / (area + areap[j] - inter);
            if (j == 0) iou0 = iou;
            bool ideq = (idsp[j] == myid);
            if (ideq && !hasm) { hasm = true; midx = j; ioum = iou; }
            float ex = ideq ? -1.0f : iou;
            if (ex > best) { best = ex; bidx = j; }
        }
        if (!hasm) ioum = iou0;   // jnp argmax of all-false eq row is 0

        const float s  = scores[t];
        const float fl = flags[t];
        const bool  f1 = (fl == 1.0f);
        const bool  hasbest = (best > 0.0f);
        const float ig1 = 1.0f - fmaxf(best, 0.0f);
        const float ig0 = 1.0f - ioum;

        const float val_cur = f1 ? (ig1 * s) : (ig0 * s * (1.0f - fl));
        const int   pj   = f1 ? bidx : midx;
        const bool  pact = f1 ? hasbest : hasm;
        const float sp = scores_prev[pj];
        const float fp = flags_prev[pj];
        const float pv = f1 ? (ig1 * ((fp == 1.0f) ? sp : sp * (1.0f - fp)))
                            : (ig0 * ((fp == 1.0f) ? sp : sp * (1.0f - fl)));

        int* r = rects + t * RECT_DW;       // current-box rect, always active
        r[0] = c0; r[1] = c1; r[2] = c2; r[3] = c3;
        r[4] = __float_as_int(val_cur); r[5] = 0; r[6] = 0; r[7] = 0;

        pj_s[t] = pj; pv_s[t] = pv; pact_s[t] = pact ? 1 : 0;
        if (pact) matched[pj] = 1;          // idempotent store, race-free
    }
    __syncthreads();

    if (t < N_CUR) {                        // associated-previous rects
        int* r = rects + (N_CUR + t) * RECT_DW;
        if (pact_s[t]) {
            const int pj = pj_s[t];
            r[0] = bp[pj][0]; r[1] = bp[pj][1]; r[2] = bp[pj][2]; r[3] = bp[pj][3];
            r[4] = __float_as_int(pv_s[t]);
        } else { r[0] = 0; r[1] = 0; r[2] = 0; r[3] = 0; r[4] = 0; }
        r[5] = 0; r[6] = 0; r[7] = 0;
    }
    for (int j = t; j < N_PREV; j += 256) { // unmatched-previous rects
        int* r = rects + (2 * N_CUR + j) * RECT_DW;
        if (!matched[j]) {
            const float sp = scores_prev[j];
            const float fp = flags_prev[j];
            const float v  = (fp == 1.0f) ? sp : sp * (1.0f - fp);
            r[0] = bp[j][0]; r[1] = bp[j][1]; r[2] = bp[j][2]; r[3] = bp[j][3];
            r[4] = __float_as_int(v);
        } else { r[0] = 0; r[1] = 0; r[2] = 0; r[3] = 0; r[4] = 0; }
        r[5] = 0; r[6] = 0; r[7] = 0;
    }
}

// ---------------------------------------------------------------------------
// Kernel 2: tiled max-paint + 2x nearest upsample, HBM-store-bound.
// Tile = 32 half-res rows x 64 half-res cols (32KB output / 13KB staged table
// per block). Rect table staged to LDS via CDNA5 async global->LDS (ASYNCcnt),
// then culled per tile; each thread owns 2 half-res cols x 4 strided rows and
// emits aligned float4 stores for the 2x-upsampled output.
// ---------------------------------------------------------------------------
__global__ __launch_bounds__(256) void ig_paint_kernel(
    const int* __restrict__ rects, float* __restrict__ out)
{
    __shared__ __align__(16) int raw[K_RECT * RECT_DW];
    __shared__ int   cx1[K_RECT], cy1[K_RECT], cx2[K_RECT], cy2[K_RECT];
    __shared__ float cval[K_RECT];
    __shared__ int   cnt;

    const int t = threadIdx.x;
    if (t == 0) cnt = 0;

    // Async-stage the 13KB rect table into LDS (832 x 16B chunks).
    const uint32_t lds_base = (uint32_t)(uintptr_t)(void*)raw; // low 32b = LDS offset
    const uint64_t gbase    = (uint64_t)(uintptr_t)rects;
    #pragma unroll
    for (int c = 0; c < 4; ++c) {
        const int chunk = t + c * 256;
        if (chunk < N_CHUNK) {
            const uint32_t voff = (uint32_t)(chunk * 16);
            const uint32_t ldst = lds_base + (uint32_t)(chunk * 16);
            asm volatile("global_load_async_to_lds_b128 %0, %1, %2"
                         :: "v"(ldst), "v"(voff), "s"(gbase) : "memory");
        }
    }
    asm volatile("s_wait_asynccnt 0" ::: "memory");
    __syncthreads();   // make all waves' async LDS writes visible

    // Cull rects against this tile, compacting survivors.
    const int tx0 = blockIdx.x * TILE_W;   // half-res tile origin
    const int ty0 = blockIdx.y * TILE_H;
    for (int r = t; r < K_RECT; r += 256) {
        const int x1 = raw[r * RECT_DW + 0];
        const int y1 = raw[r * RECT_DW + 1];
        const int x2 = raw[r * RECT_DW + 2];
        const int y2 = raw[r * RECT_DW + 3];
        if (x1 < tx0 + TILE_W && x2 > tx0 && y1 < ty0 + TILE_H && y2 > ty0) {
            const int p = atomicAdd(&cnt, 1);
            cx1[p] = x1; cy1[p] = y1; cx2[p] = x2; cy2[p] = y2;
            cval[p] = __int_as_float(raw[r * RECT_DW + 4]);
        }
    }
    __syncthreads();
    const int n = cnt;

    // Thread footprint: 2 adjacent half-res cols, 4 rows strided by 8.
    const int tx = t & 31;                 // 32 threads x 2 cols = 64 cols
    const int ty = t >> 5;                 // 8 base rows
    const int xh = tx0 + tx * 2;
    const size_t col4 = (size_t)(xh * 2) >> 2;   // float4 col index (16B aligned)
    float4* o = (float4*)out;

    #pragma unroll
    for (int rr = 0; rr < ROWS_PER_THREAD; ++rr) {
        const int yh = ty0 + ty + rr * ROW_STRIDE;

        float v0 = 0.0f, v1 = 0.0f;
        for (int k = 0; k < n; ++k) {
            const int b1 = cy1[k], b2 = cy2[k];
            if (yh >= b1 && yh < b2) {
                const int a1 = cx1[k], a2 = cx2[k];
                const float v = cval[k];
                if (xh     >= a1 && xh     < a2) v0 = fmaxf(v0, v);
                if (xh + 1 >= a1 && xh + 1 < a2) v1 = fmaxf(v1, v);
            }
        }

        if (yh < HS) {
            const float4 q = make_float4(v0, v0, v1, v1);
            o[(size_t)(2 * yh)     * (W_FULL / 4) + col4] = q;
            o[(size_t)(2 * yh + 1) * (W_FULL / 4) + col4] = q;
        }
    }
}

// ---------------------------------------------------------------------------
extern "C" void kernel_launch(void* const* d_in, const int* in_sizes, int n_in,
                              void* d_out, int out_size, void* d_ws, size_t ws_size,
                              hipStream_t stream) {
    (void)in_sizes; (void)n_in; (void)out_size; (void)ws_size;
    // setup_inputs order: inputs(unused), boxes, scores, flags,
    //                     boxes_prev, scores_prev, flags_prev, ids, ids_prev
    const float* boxes       = (const float*)d_in[1];
    const float* scores      = (const float*)d_in[2];
    const float* flags       = (const float*)d_in[3];
    const float* boxes_prev  = (const float*)d_in[4];
    const float* scores_prev = (const float*)d_in[5];
    const float* flags_prev  = (const float*)d_in[6];
    const int*   ids         = (const int*)d_in[7];
    const int*   ids_prev    = (const int*)d_in[8];
    int*   rects = (int*)d_ws;          // K_RECT * 8 dwords = 13312 bytes
    float* out   = (float*)d_out;       // [1,1,2160,3840] f32

    ig_prep_kernel<<<dim3(1), dim3(256), 0, stream>>>(
        boxes, scores, flags, boxes_prev, scores_prev, flags_prev,
        ids, ids_prev, rects);

    ig_paint_kernel<<<dim3(WSUB / TILE_W, (HS + TILE_H - 1) / TILE_H), dim3(256), 0, stream>>>(
        rects, out);
}